// GNNModel_90555090469520
// MI455X (gfx1250) — compile-verified
//
#include <hip/hip_runtime.h>
#include <hip/hip_bf16.h>
#include <stdint.h>

// ---------------------------------------------------------------------------
// GCN forward on MI455X (gfx1250): fp32 WMMA GEMM + L2-resident edge scatter
// with hardware device-scope f32 atomics via inline CDNA5 asm.
// ---------------------------------------------------------------------------

typedef float v2f __attribute__((ext_vector_type(2)));
typedef float v8f __attribute__((ext_vector_type(8)));

// Hardware fp32 atomic add, no return (STOREcnt-tracked), device scope so the
// RMW happens at the device-coherent point (L2). Guarantees we never fall back
// to a CAS loop on the dominant scatter kernel.
__device__ __forceinline__ void atomic_add_f32_dev(float* p, float v) {
    asm volatile("global_atomic_add_f32 %0, %1, off scope:SCOPE_DEV"
                 :: "v"(p), "v"(v)
                 : "memory");
}

// ---------------- utility ----------------
__global__ void fill_f32(float* __restrict__ p, float v, long long nelem) {
    long long i = (long long)blockIdx.x * blockDim.x + threadIdx.x;
    if (i < nelem) p[i] = v;
}

// ---------------- GCN normalization ----------------
__global__ void degree_kernel(const int* __restrict__ dst, float* __restrict__ deg, int E) {
    int e = blockIdx.x * blockDim.x + threadIdx.x;
    if (e < E) atomic_add_f32_dev(&deg[dst[e]], 1.0f);
}

__global__ void rsqrt_kernel(float* __restrict__ dinv, int n) {
    int i = blockIdx.x * blockDim.x + threadIdx.x;
    if (i < n) {
        float d = dinv[i];
        dinv[i] = d > 0.0f ? rsqrtf(d) : 0.0f;
    }
}

__global__ void norm_kernel(const int* __restrict__ src, const int* __restrict__ dst,
                            const float* __restrict__ dinv, float* __restrict__ nrm, int E) {
    int e = blockIdx.x * blockDim.x + threadIdx.x;
    if (e < E) nrm[e] = dinv[src[e]] * dinv[dst[e]];
}

// ---------------- fp32 WMMA GEMM:  H[M,F] = X[M,K] * W[K,F] ----------------
// Block = 128 threads (4 waves). One 16-row A tile staged in LDS (stride K+4
// keeps the 16-lane column reads on distinct banks given 64 banks). Each wave
// owns TWO 16x16 N-tiles (all widths are multiples of 32): one A-fragment read
// feeds two independent V_WMMA_F32_16X16X4_F32 chains, halving LDS traffic per
// FLOP and breaking the single-accumulator RAW chain.
//
// Fragment layouts per ISA 7.12.2 (wave32):
//   A 16x4:  lanes 0-15 row M=lane, v0=K0 v1=K1; lanes 16-31 same rows, K2/K3.
//   B 4x16:  lanes 0-15 col N=lane, rows k+0/k+1 in v0/v1; lanes 16-31 k+2/k+3.
//   C/D:     VGPR v holds (M = v + 8*(lane>=16), N = lane&15).
__global__ __launch_bounds__(128) void gemm_wmma_f32(
        const float* __restrict__ X, const float* __restrict__ W,
        float* __restrict__ H, int M, int K, int N) {
    __shared__ float As[16 * (512 + 4)];
    const int ldA = K + 4;
    const int m0  = blockIdx.x * 16;
    const int tid = threadIdx.x;

    // cooperative, coalesced load of the 16 x K A tile
    for (int r = 0; r < 16; ++r) {
        const int row = m0 + r;
        for (int c = tid; c < K; c += 128) {
            As[r * ldA + c] = (row < M) ? X[(size_t)row * K + c] : 0.0f;
        }
    }
    __syncthreads();

    const int lane = tid & 31;
    const int wv   = tid >> 5;
    const int n0   = (blockIdx.y * 4 + wv) * 32;   // two 16-wide tiles per wave
    if (n0 >= N) return;                           // wave-uniform guard

    const int half = lane >> 4;                    // 0: lanes 0-15, 1: lanes 16-31
    const int l16  = lane & 15;
    const float* Wc0 = W + n0 + l16;               // tile 0 column base
    const float* Wc1 = Wc0 + 16;                   // tile 1 column base

    v8f acc0 = {};
    v8f acc1 = {};
    for (int k = 0; k < K; k += 4) {
        const int kk = k + half * 2;
        v2f a, b0, b1;
        a.x  = As[l16 * ldA + kk];
        a.y  = As[l16 * ldA + kk + 1];
        b0.x = Wc0[(size_t)(kk)     * N];
        b0.y = Wc0[(size_t)(kk + 1) * N];
        b1.x = Wc1[(size_t)(kk)     * N];
        b1.y = Wc1[(size_t)(kk + 1) * N];
        acc0 = __builtin_amdgcn_wmma_f32_16x16x4_f32(
                   false, a, false, b0, (short)0, acc0, false, false);
        acc1 = __builtin_amdgcn_wmma_f32_16x16x4_f32(
                   false, a, false, b1, (short)0, acc1, false, false);
    }

    for (int v = 0; v < 8; ++v) {
        const int row = m0 + v + half * 8;
        if (row < M) {
            float* hp = H + (size_t)row * N + n0 + l16;
            hp[0]  = acc0[v];
            hp[16] = acc1[v];
        }
    }
}

// ---------------- aggregation ----------------
// agg = H * dinv^2  (self-loop term; also (re)initializes the poisoned buffer)
__global__ void selfloop_kernel(const float* __restrict__ H, const float* __restrict__ dinv,
                                float* __restrict__ agg, int n, int F) {
    long long t = (long long)blockIdx.x * blockDim.x + threadIdx.x;
    long long total = (long long)n * F;
    if (t >= total) return;
    int i = (int)(t / F);
    float d = dinv[i];
    agg[t] = H[t] * d * d;
}

// agg[dst] += H[src] * norm   — wave covers consecutive float4 chunks of one
// edge => gather of a source row is fully coalesced; adds are hardware L2
// atomics (global_atomic_add_f32 scope:SCOPE_DEV).
__global__ void scatter_kernel(const float* __restrict__ H, const float* __restrict__ nrm,
                               const int* __restrict__ src, const int* __restrict__ dst,
                               float* __restrict__ agg, int F, int chunks, int E) {
    long long t = (long long)blockIdx.x * blockDim.x + threadIdx.x;
    long long total = (long long)E * chunks;
    if (t >= total) return;
    int e = (int)(t / chunks);
    int c = (int)(t % chunks);
    int s = src[e], d = dst[e];
    float w = nrm[e];
    const float4 v = *((const float4*)(H + (size_t)s * F) + c);
    float* ap = agg + (size_t)d * F + (size_t)c * 4;
    atomic_add_f32_dev(ap + 0, v.x * w);
    atomic_add_f32_dev(ap + 1, v.y * w);
    atomic_add_f32_dev(ap + 2, v.z * w);
    atomic_add_f32_dev(ap + 3, v.w * w);
}

// in-place bias + activation  (mode 0=relu, 1=leaky(0.01), 2=none)
__global__ void bias_act_kernel(float* __restrict__ agg, const float* __restrict__ b,
                                int n, int F, int mode) {
    long long t = (long long)blockIdx.x * blockDim.x + threadIdx.x;
    if (t >= (long long)n * F) return;
    int f = (int)(t % F);
    float v = agg[t] + b[f];
    if (mode == 0)      v = v > 0.0f ? v : 0.0f;
    else if (mode == 1) v = v > 0.0f ? v : 0.01f * v;
    agg[t] = v;
}

// ---------------- global mean pool ----------------
__global__ void pool_sum_kernel(const float* __restrict__ h, const int* __restrict__ batch,
                                float* __restrict__ out, float* __restrict__ counts, int n) {
    long long t = (long long)blockIdx.x * blockDim.x + threadIdx.x;
    if (t >= (long long)n * 32) return;
    int i = (int)(t >> 5);
    int f = (int)(t & 31);
    int g = batch[i];
    atomic_add_f32_dev(&out[g * 32 + f], h[t]);
    if (f == 0) atomic_add_f32_dev(&counts[g], 1.0f);
}

__global__ void pool_div_kernel(float* __restrict__ out, const float* __restrict__ counts,
                                int total) {
    int t = blockIdx.x * blockDim.x + threadIdx.x;
    if (t < total) out[t] /= fmaxf(counts[t >> 5], 1.0f);
}

// ---------------------------------------------------------------------------
static inline int cdiv_ll(long long a, long long b) { return (int)((a + b - 1) / b); }

extern "C" void kernel_launch(void* const* d_in, const int* in_sizes, int n_in,
                              void* d_out, int out_size, void* d_ws, size_t ws_size,
                              hipStream_t stream) {
    const float* x     = (const float*)d_in[0];
    const int*   ei    = (const int*)d_in[1];   // [2, E]
    const int*   batch = (const int*)d_in[2];   // [n]
    const float* Ws[9];
    const float* bs[9];
    for (int i = 0; i < 9; ++i) {
        Ws[i] = (const float*)d_in[3 + 2 * i];
        bs[i] = (const float*)d_in[4 + 2 * i];
    }
    const int E = in_sizes[1] / 2;
    const int n = in_sizes[2];
    const int* src = ei;
    const int* dst = ei + E;

    // workspace carve-out (floats): dinv | norm | counts | bufH | bufAgg
    float* ws     = (float*)d_ws;
    float* dinv   = ws;                         // n (<= 50016)
    float* nrm    = ws + 50016;                 // E
    float* counts = nrm + E;                    // 64
    float* bufH   = counts + 64;                // n * 512 max
    float* bufG   = bufH + (size_t)50000 * 512; // n * 512 max
    (void)ws_size;

    const int T = 256;

    // --- symmetric normalization (self-loop folded into deg init = 1.0) ---
    fill_f32<<<cdiv_ll(n, T), T, 0, stream>>>(dinv, 1.0f, n);
    degree_kernel<<<cdiv_ll(E, T), T, 0, stream>>>(dst, dinv, E);
    rsqrt_kernel<<<cdiv_ll(n, T), T, 0, stream>>>(dinv, n);
    norm_kernel<<<cdiv_ll(E, T), T, 0, stream>>>(src, dst, dinv, nrm, E);

    static const int widths[10] = {128, 128, 256, 384, 512, 512, 384, 256, 128, 32};
    static const int actm[9]    = {0, 0, 1, 0, 1, 1, 0, 0, 2};

    const float* cur = x;
    for (int l = 0; l < 9; ++l) {
        const int K = widths[l];
        const int F = widths[l + 1];
        // dense transform: bufH = cur @ W   (4 waves/block, 32 cols/wave)
        dim3 grid(cdiv_ll(n, 16), cdiv_ll(F, 128));
        gemm_wmma_f32<<<grid, 128, 0, stream>>>(cur, Ws[l], bufH, n, K, F);
        // aggregate: bufG = D^-1/2 (A+I) D^-1/2 @ bufH
        long long nf = (long long)n * F;
        selfloop_kernel<<<cdiv_ll(nf, T), T, 0, stream>>>(bufH, dinv, bufG, n, F);
        const int chunks = F / 4;
        long long tot = (long long)E * chunks;
        scatter_kernel<<<cdiv_ll(tot, T), T, 0, stream>>>(bufH, nrm, src, dst, bufG, F, chunks, E);
        // bias + activation in place; bufG becomes next layer's input
        bias_act_kernel<<<cdiv_ll(nf, T), T, 0, stream>>>(bufG, bs[l], n, F, actm[l]);
        cur = bufG;
    }

    // --- global mean pool over 64 graphs (F = 32) ---
    float* out = (float*)d_out;
    fill_f32<<<cdiv_ll(out_size, T), T, 0, stream>>>(out, 0.0f, out_size);
    fill_f32<<<1, 64, 0, stream>>>(counts, 0.0f, 64);
    pool_sum_kernel<<<cdiv_ll((long long)n * 32, T), T, 0, stream>>>(bufG, batch, out, counts, n);
    pool_div_kernel<<<cdiv_ll(out_size, T), T, 0, stream>>>(out, counts, out_size);
}